// Const2DepCRF_33904471834773
// MI455X (gfx1250) — compile-verified
//
#include <hip/hip_runtime.h>
#include <stdint.h>

// Eisner inside algorithm (logsumexp semiring), two passes
// (unconstrained logZ, constrained score), loss = (logZ - score)/tokens.
// One workgroup per (batch, pass). All DP state LDS-resident (304KB/WGP,
// CDNA5-only). Arc/mask tiles DMA'd into LDS via the Tensor Data Mover,
// with the per-width combine_mask slice double-buffered so the DMA overlaps
// phase-2(w) and phase-1(w+1) compute.

#define SDIM 128
#define MINF (-1e32f)
#define NEGBIG (-3.0e38f)

typedef unsigned int v4u __attribute__((ext_vector_type(4)));
typedef int          v8i __attribute__((ext_vector_type(8)));
typedef int          v4i __attribute__((ext_vector_type(4)));

// ---- Tensor Data Mover: 2D tile load global -> LDS --------------------------
// Descriptor layout per CDNA5 ISA ch.8:
//  group0: [1:0]=count(1), [63:32]=lds_addr, [120:64]=global_addr, [127:126]=type(2)
//  group1: [17:16]=data_size, [79:48]=tensor_dim0, [111:80]=tensor_dim1,
//          [127:112]=tile_dim0, [143:128]=tile_dim1, [207:160]=tensor_dim0_stride
__device__ __forceinline__ void tdm_load_2d(uint32_t lds_off, const void* gptr,
                                            uint32_t dsize_log2, uint32_t dim0,
                                            uint32_t dim1, uint32_t stride0) {
  uint64_t ga = (uint64_t)(uintptr_t)gptr;
  v4u g0;
  g0[0] = 1u;                                                // count=1, user mode
  g0[1] = lds_off;                                           // LDS byte address
  g0[2] = (uint32_t)(ga & 0xffffffffu);                      // global_addr[31:0]
  g0[3] = (uint32_t)((ga >> 32) & 0x01ffffffu) | (2u << 30); // addr[56:32], type=2
  v8i g1;
  g1[0] = (int)(dsize_log2 << 16);                           // wg_mask=0, data_size
  g1[1] = (int)((dim0 & 0xffffu) << 16);                     // tensor_dim0[15:0]
  g1[2] = (int)(((dim0 >> 16) & 0xffffu) | ((dim1 & 0xffffu) << 16));
  g1[3] = (int)(((dim1 >> 16) & 0xffffu) | ((dim0 & 0xffffu) << 16)); // tile_dim0
  g1[4] = (int)(dim1 & 0xffffu);                             // tile_dim1, tile_dim2=0
  g1[5] = (int)stride0;                                      // dim0_stride[31:0]
  g1[6] = 0;
  g1[7] = 0;
  v4i gz = {0, 0, 0, 0};
#if defined(__clang_major__) && (__clang_major__ >= 23)
  v8i gz8 = {0, 0, 0, 0, 0, 0, 0, 0};
  __builtin_amdgcn_tensor_load_to_lds(g0, g1, gz, gz, gz8, 0);
#else
  __builtin_amdgcn_tensor_load_to_lds(g0, g1, gz, gz, 0);
#endif
}

// ---- wave32 reductions ------------------------------------------------------
__device__ __forceinline__ float wred_max(float v) {
#pragma unroll
  for (int o = 16; o > 0; o >>= 1) v = fmaxf(v, __shfl_xor(v, o, 32));
  return v;
}
__device__ __forceinline__ float wred_sum(float v) {
#pragma unroll
  for (int o = 16; o > 0; o >>= 1) v += __shfl_xor(v, o, 32);
  return v;
}

// ---- main DP kernel: one workgroup per (batch, pass) ------------------------
__global__ __launch_bounds__(512) void eisner_inside_kernel(
    const float* __restrict__ scores,        // [B,S,S]
    const unsigned char* __restrict__ mask,  // [B,S]
    const unsigned char* __restrict__ cspan, // [B,S,S]   True = span allowed
    const unsigned char* __restrict__ comb,  // [B,S,S,S] True = split forbidden
    float* __restrict__ part)                // [2*B] root scores
{
  extern __shared__ unsigned char smem[];
  const int S = SDIM;
  float* sc   = (float*)(smem);                      // s_c row-major      64KB
  float* scT  = (float*)(smem + 1 * S * S * 4);      // s_c transposed     64KB
  float* si   = (float*)(smem + 2 * S * S * 4);      // s_i row-major      64KB
  float* arc  = (float*)(smem + 3 * S * S * 4);      // scores[b] tile     64KB
  unsigned char* csm  = smem + 4 * S * S * 4;        // c_span_mask[b]     16KB
  unsigned char* cmb0 = smem + 4 * S * S * 4 + S*S;  // combine slice buf0 16KB
  unsigned char* cmb1 = cmb0 + S * S;                // combine slice buf1 16KB
  __shared__ int s_len;

  const int b = blockIdx.x;
  const bool constrained = (blockIdx.y == 1);
  const int t = threadIdx.x;
  const int lane = t & 31;
  const int wave = t >> 5;
  const int nwaves = blockDim.x >> 5;
  const unsigned char* comb_b = comb + (size_t)b * S * (size_t)(S * S);

  // init s_c (diag 0, else MIN) in both layouts
  for (int i = t; i < S * S; i += blockDim.x) {
    float v = ((i / S) == (i % S)) ? 0.0f : MINF;
    sc[i] = v;
    scT[i] = v;
  }
  // DMA arc tile (f32 128x128) and span-mask tile (u8 128x128) into LDS,
  // then prefetch the first combine_mask slice (w=1) into buffer 1.
  if (wave == 0) {
    tdm_load_2d((uint32_t)(uintptr_t)arc, scores + (size_t)b * S * S,
                /*4B*/ 2u, S, S, S);
    if (constrained) {
      tdm_load_2d((uint32_t)(uintptr_t)csm, cspan + (size_t)b * S * S,
                  /*1B*/ 0u, S, S, S);
      tdm_load_2d((uint32_t)(uintptr_t)cmb1, comb_b + (size_t)(S * S),
                  /*1B*/ 0u, S, S, S);
    }
  }
  if (t == 0) {
    int L = 0;
    for (int s2 = 0; s2 < S; ++s2) L += mask[b * S + s2] ? 1 : 0;
    s_len = L;
  }
  if (wave == 0) __builtin_amdgcn_s_wait_tensorcnt(0);  // arc (+csm) ready
  __syncthreads();
  const int len = s_len;

  for (int w = 1; w < S; ++w) {
    const int n = S - w;
    unsigned char* cmb = (w & 1) ? cmb1 : cmb0;

    // ---- phase 1: incomplete spans  I(i,j) -------------------------------
    for (int k = wave; k < n; k += nwaves) {
      float v[4];
      float m = NEGBIG;
#pragma unroll
      for (int j = 0; j < 4; ++j) {
        int l = lane + 32 * j;
        int lc = (l < w) ? l : 0;
        float x = sc[k * S + (k + lc)] + sc[(w + k) * S + (1 + k + lc)];
        v[j] = (l < w) ? x : NEGBIG;
        m = fmaxf(m, v[j]);
      }
      m = wred_max(m);
      float s = 0.0f;
#pragma unroll
      for (int j = 0; j < 4; ++j) s += __expf(v[j] - m);
      s = wred_sum(s);
      if (lane == 0) {
        float il = m + __logf(s);
        // s_arc[x,y] = scores[b,y,x] ; arc tile holds scores[b] row-major
        si[(w + k) * S + k] = il + arc[k * S + (w + k)];
        si[k * S + (w + k)] = il + arc[(w + k) * S + k];
      }
    }
    if (constrained && wave == 0)
      __builtin_amdgcn_s_wait_tensorcnt(0);  // combine slice for width w ready
    __syncthreads();

    // prefetch combine slice for width w+1 into the other buffer; the DMA
    // overlaps phase 2 of width w and phase 1 of width w+1.
    if (constrained && wave == 0 && (w + 1) < S)
      tdm_load_2d((uint32_t)(uintptr_t)((w & 1) ? cmb0 : cmb1),
                  comb_b + (size_t)(w + 1) * (size_t)(S * S), 0u, S, S, S);

    // ---- phase 2: complete spans  C(j->i), C(i->j) -----------------------
    for (int k = wave; k < n; k += nwaves) {
      float vl[4], vr[4];
      float ml = NEGBIG, mr = NEGBIG;
#pragma unroll
      for (int j = 0; j < 4; ++j) {
        int l = lane + 32 * j;
        int lc = (l < w) ? l : 0;
        // cl = s_c[k+l, k] + s_i[w+k, k+l]   (column read via scT)
        float cl = scT[k * S + (k + lc)] + si[(w + k) * S + (k + lc)];
        // cr = s_i[k, 1+k+l] + s_c[1+k+l, w+k]
        float cr = si[k * S + (1 + k + lc)] + scT[(w + k) * S + (1 + k + lc)];
        if (constrained) {
          if (cmb[(k + lc) * S + k]) cl = MINF;
          if (cmb[(1 + k + lc) * S + (w + k)]) cr = MINF;
        }
        vl[j] = (l < w) ? cl : NEGBIG;
        vr[j] = (l < w) ? cr : NEGBIG;
        ml = fmaxf(ml, vl[j]);
        mr = fmaxf(mr, vr[j]);
      }
      ml = wred_max(ml);
      mr = wred_max(mr);
      float sl = 0.0f, sr = 0.0f;
#pragma unroll
      for (int j = 0; j < 4; ++j) {
        sl += __expf(vl[j] - ml);
        sr += __expf(vr[j] - mr);
      }
      sl = wred_sum(sl);
      sr = wred_sum(sr);
      if (lane == 0) {
        float L = ml + __logf(sl);
        float R = mr + __logf(sr);
        if (constrained) {
          if (!csm[k * S + (w + k)]) L = MINF;        // c_span_mask[b,k,w+k]
          if (!csm[(w + k) * S + k]) R = MINF;        // c_span_mask[b,w+k,k]
        }
        if (k == 0) R = (len == w) ? R : MINF;        // single-root constraint
        sc[(w + k) * S + k] = L;
        scT[k * S + (w + k)] = L;
        sc[k * S + (w + k)] = R;
        scT[(w + k) * S + k] = R;
      }
    }
    __syncthreads();
  }
  if (t == 0) part[blockIdx.y * gridDim.x + b] = sc[len];  // s_c[0][len]
}

// ---- finalize: loss = (logZ - score) / tokens -------------------------------
__global__ void eisner_finalize_kernel(const unsigned char* __restrict__ mask,
                                       const float* __restrict__ part,
                                       float* __restrict__ out, int B) {
  const int S = SDIM;
  int lane = threadIdx.x;
  float lz = 0.0f, sc = 0.0f;
  int tok = 0;
  for (int b = lane; b < B; b += 32) {
    lz += part[b];
    sc += part[B + b];
    for (int s2 = 0; s2 < S; ++s2) tok += mask[b * S + s2] ? 1 : 0;
  }
#pragma unroll
  for (int o = 16; o > 0; o >>= 1) {
    lz += __shfl_xor(lz, o, 32);
    sc += __shfl_xor(sc, o, 32);
    tok += __shfl_xor(tok, o, 32);
  }
  if (lane == 0) out[0] = (lz - sc) / (float)tok;
}

// ---- marginals passthrough: out[1..] = scores ------------------------------
__global__ void eisner_copy_kernel(const float* __restrict__ scores,
                                   float* __restrict__ out, int n) {
  for (int i = blockIdx.x * blockDim.x + threadIdx.x; i < n;
       i += gridDim.x * blockDim.x)
    out[i] = scores[i];
}

extern "C" void kernel_launch(void* const* d_in, const int* in_sizes, int n_in,
                              void* d_out, int out_size, void* d_ws, size_t ws_size,
                              hipStream_t stream) {
  (void)n_in; (void)out_size; (void)ws_size;
  const float* scores        = (const float*)d_in[0];
  const unsigned char* mask  = (const unsigned char*)d_in[1];
  const unsigned char* cspan = (const unsigned char*)d_in[2];
  const unsigned char* comb  = (const unsigned char*)d_in[3];
  float* out  = (float*)d_out;
  float* part = (float*)d_ws;                 // 2*B floats of scratch

  const int S = SDIM;
  const int B = in_sizes[1] / S;              // mask is [B,S]
  // sc + scT + si + arc (4 x 64KB) + csm (16KB) + 2 combine slice bufs (32KB)
  const size_t shmem = (size_t)4 * S * S * sizeof(float) + (size_t)3 * S * S; // 304KB

  hipLaunchKernelGGL(eisner_inside_kernel, dim3(B, 2), dim3(512), shmem, stream,
                     scores, mask, cspan, comb, part);
  hipLaunchKernelGGL(eisner_finalize_kernel, dim3(1), dim3(32), 0, stream,
                     mask, part, out, B);
  hipLaunchKernelGGL(eisner_copy_kernel, dim3(1024), dim3(256), 0, stream,
                     scores, out + 1, in_sizes[0]);
}